// DGCNN_18657337933834
// MI455X (gfx1250) — compile-verified
//
#include <hip/hip_runtime.h>
#include <hip/hip_bf16.h>
#include <math.h>

// ---------------- problem constants (match reference) ----------------
#define N_NODES 131072
#define NPG     64
#define G_GRAPHS 2048
#define E_EDGES 2097152
#define HDIM    128
#define KPOOL   30
#define C1_CH   32
#define KS_     5
#define TCONV   26          // KPOOL - KS + 1
#define FC1_K   832         // C1_CH * TCONV
#define NCLS    2

typedef __attribute__((ext_vector_type(2))) float v2f;
typedef __attribute__((ext_vector_type(8))) float v8f;

// CDNA5 fp32 WMMA: D(16x16) = A(16x4) * B(4x16) + C   (wave32, 8-arg builtin)
__device__ __forceinline__ v8f wmma_f32x4(v2f a, v2f b, v8f c) {
    return __builtin_amdgcn_wmma_f32_16x16x4_f32(
        /*neg_a=*/false, a, /*neg_b=*/false, b,
        /*c_mod=*/(short)0, c, /*reuse_a=*/false, /*reuse_b=*/false);
}

// ---------------- degree / dinv ----------------
__global__ void deg_init_kernel(float* deg) {
    int i = blockIdx.x * blockDim.x + threadIdx.x;
    if (i < N_NODES) deg[i] = 1.0f;               // self loop
}
__global__ void deg_count_kernel(const int* ei, float* deg) {
    int e = blockIdx.x * blockDim.x + threadIdx.x;
    if (e < E_EDGES) atomicAdd(&deg[ei[E_EDGES + e]], 1.0f);   // dst row
}
__global__ void dinv_kernel(float* deg) {
    int i = blockIdx.x * blockDim.x + threadIdx.x;
    if (i < N_NODES) deg[i] = rsqrtf(deg[i]);
}

// ---------------- weight pre-pack: B[K,N] -> pair rows Wp[k/2][n] = {B[2k2][n], B[2k2+1][n]} ----------------
__global__ void pack_b_kernel(const float* __restrict__ B, v2f* __restrict__ out, int K, int N) {
    int idx = blockIdx.x * blockDim.x + threadIdx.x;      // (K/2)*N threads
    if (idx >= (K >> 1) * N) return;
    int k2 = idx / N, n = idx % N;
    v2f p = { B[(size_t)(2 * k2) * N + n], B[(size_t)(2 * k2 + 1) * N + n] };
    out[idx] = p;
}

// conv weights: logical B[k = s*128+h][c] = convw[c,h,s]; pack pairs (h even, h+1) directly
__global__ void convw_pack_kernel(const float* __restrict__ convw, v2f* __restrict__ wTp) {
    int idx = blockIdx.x * blockDim.x + threadIdx.x;      // (640/2)*32 = 10240
    if (idx >= (KS_ * HDIM / 2) * C1_CH) return;
    int k2 = idx / C1_CH, c = idx % C1_CH;
    int k = 2 * k2;
    int s = k / HDIM, h = k % HDIM;                       // k even -> h even, h+1 <= 127
    v2f p = { convw[(size_t)c * (HDIM * KS_) + h * KS_ + s],
              convw[(size_t)c * (HDIM * KS_) + (h + 1) * KS_ + s] };
    wTp[idx] = p;
}

// ---------------- generic fp32 WMMA GEMM: C[M,N] = act(A[M,K]) @ B[K,N] (+bias)(+relu) ----------------
// B pre-packed as pair rows (v2f). One wave per 16x16 tile; tiles must divide waves/block.
template <bool ACTRELU>
__global__ void gemm_wmma_kernel(const float* __restrict__ A, const v2f* __restrict__ Bp2,
                                 const float* __restrict__ bias, float* __restrict__ C,
                                 int M, int N, int K, int relu) {
    const int lane  = threadIdx.x & 31;
    const int wave  = threadIdx.x >> 5;
    const int wpb   = blockDim.x >> 5;
    const int tilesN = N >> 4;
    const int numTiles = (M >> 4) * tilesN;
    int tileId = blockIdx.x * wpb + wave;
    if (tileId >= numTiles) return;               // wave-uniform
    const int tm = tileId / tilesN, tn = tileId % tilesN;
    const int l15  = lane & 15;
    const int half = lane >> 4;                   // 0: K pair {0,1}, 1: K pair {2,3}
    const float* Ap = A   + (size_t)(tm * 16 + l15) * K + half * 2;
    const v2f*   Bp = Bp2 + (size_t)half * N + (tn * 16 + l15);
    v8f acc = {};
    for (int kc = 0; kc < K; kc += 32) {          // K is a multiple of 32 for all call sites
        __builtin_prefetch(Ap + kc + 32, 0, 1);   // global_prefetch_b8 (speculative)
#pragma unroll
        for (int kk = 0; kk < 32; kk += 4) {
            const int k = kc + kk;
            v2f a = { Ap[k], Ap[k + 1] };
            if (ACTRELU) { a.x = fmaxf(a.x, 0.0f); a.y = fmaxf(a.y, 0.0f); }
            v2f b = Bp[(size_t)(k >> 1) * N];
            acc = wmma_f32x4(a, b, acc);
        }
    }
    const int ccol = tn * 16 + l15;
    const int crow = tm * 16 + half * 8;
#pragma unroll
    for (int v = 0; v < 8; ++v) {
        float r = acc[v];
        if (bias) r += bias[ccol];
        if (relu) r = fmaxf(r, 0.0f);
        C[(size_t)(crow + v) * N + ccol] = r;
    }
}

// ---------------- GCN per-layer elementwise + scatter ----------------
__global__ void self_bias_kernel(const float* __restrict__ hw, const float* __restrict__ dinv,
                                 const float* __restrict__ bias, float* __restrict__ out) {
    size_t idx = (size_t)blockIdx.x * blockDim.x + threadIdx.x;   // N*128 threads
    int node = (int)(idx >> 7), c = (int)(idx & 127);
    float d = dinv[node];
    out[idx] = hw[idx] * d * d + bias[c];
}

__global__ void scatter_kernel(const int* __restrict__ ei, const float* __restrict__ dinv,
                               const float* __restrict__ hw, float* __restrict__ out) {
    long long gid = (long long)blockIdx.x * blockDim.x + threadIdx.x;
    int e  = (int)(gid >> 5);
    if (e >= E_EDGES) return;
    int cg = (int)(gid & 31) << 2;                // 4 channels per lane
    int s = ei[e], d = ei[E_EDGES + e];
    float norm = dinv[s] * dinv[d];
    const float4 hv = *(const float4*)(hw + (size_t)s * HDIM + cg);
    float* o = out + (size_t)d * HDIM + cg;
    atomicAdd(o + 0, hv.x * norm);
    atomicAdd(o + 1, hv.y * norm);
    atomicAdd(o + 2, hv.z * norm);
    atomicAdd(o + 3, hv.w * norm);
}

// ---------------- sort-pool: relu fused; per graph, top-30 rows by channel 127 desc (stable) ----------------
__global__ void sortpool_kernel(const float* __restrict__ h, float* __restrict__ pooled) {
    __shared__ float vals[NPG];
    int g = blockIdx.x;
    int i = threadIdx.x;                          // 0..63
    const float* hg = h + (size_t)g * NPG * HDIM;
    float v = fmaxf(hg[i * HDIM + (HDIM - 1)], 0.0f);     // post-ReLU key (matches reference)
    vals[i] = v;
    __syncthreads();
    int rank = 0;
    for (int j = 0; j < NPG; ++j) {
        float vj = vals[j];
        rank += (vj > v) || (vj == v && j < i);   // argsort(-x) stable tie-break
    }
    if (rank < KPOOL) {
        float4* dst = (float4*)(pooled + (size_t)g * KPOOL * HDIM + (size_t)rank * HDIM);
        const float4* src = (const float4*)(hg + (size_t)i * HDIM);
#pragma unroll
        for (int c4 = 0; c4 < HDIM / 4; ++c4) {
            float4 t = src[c4];
            t.x = fmaxf(t.x, 0.0f); t.y = fmaxf(t.y, 0.0f);
            t.z = fmaxf(t.z, 0.0f); t.w = fmaxf(t.w, 0.0f);
            dst[c4] = t;
        }
    }
}

// ---------------- conv1d as WMMA GEMM: rows r = g*26+t, A[r][k] = pooled[g, t*128 + k] ----------------
// store transposed to match reshape(G,-1): out[g*832 + c*26 + t]
__global__ void conv_gemm_kernel(const float* __restrict__ pooled, const v2f* __restrict__ wTp,
                                 const float* __restrict__ convb, float* __restrict__ out) {
    const int lane = threadIdx.x & 31;
    const int wave = threadIdx.x >> 5;
    int tileId = blockIdx.x * (blockDim.x >> 5) + wave;   // (G*26/16)*2 = 6656 tiles
    if (tileId >= (G_GRAPHS * TCONV / 16) * 2) return;
    const int tm = tileId >> 1, tn = tileId & 1;
    const int l15 = lane & 15, half = lane >> 4;
    const int r = tm * 16 + l15;
    const int g = r / TCONV, t = r % TCONV;
    const float* Ap = pooled + (size_t)g * (KPOOL * HDIM) + (size_t)t * HDIM + half * 2;
    const v2f*   Bp = wTp + (size_t)half * C1_CH + (tn * 16 + l15);
    v8f acc = {};
    for (int kc = 0; kc < KS_ * HDIM; kc += 32) {         // K = 640
        __builtin_prefetch(Ap + kc + 32, 0, 1);
#pragma unroll
        for (int kk = 0; kk < 32; kk += 4) {
            const int k = kc + kk;
            v2f a = { Ap[k], Ap[k + 1] };
            v2f b = Bp[(size_t)(k >> 1) * C1_CH];
            acc = wmma_f32x4(a, b, acc);
        }
    }
    const int c = tn * 16 + l15;
    const float bi = convb[c];
#pragma unroll
    for (int v = 0; v < 8; ++v) {
        int rr = tm * 16 + half * 8 + v;
        int gg = rr / TCONV, tt = rr % TCONV;
        out[(size_t)gg * FC1_K + c * TCONV + tt] = fmaxf(acc[v] + bi, 0.0f);
    }
}

// ---------------- fc3 + log_softmax (N=2, scalar) ----------------
__global__ void fc3_lsm_kernel(const float* __restrict__ fc2o, const float* __restrict__ lw3,
                               const float* __restrict__ lb3, float* __restrict__ out) {
    int g = blockIdx.x * blockDim.x + threadIdx.x;
    if (g >= G_GRAPHS) return;
    const float* a = fc2o + (size_t)g * 64;
    float z0 = lb3[0], z1 = lb3[1];
    for (int j = 0; j < 64; ++j) {
        float aj = a[j];
        z0 += aj * lw3[j * 2 + 0];
        z1 += aj * lw3[j * 2 + 1];
    }
    float m = fmaxf(z0, z1);
    float lse = m + logf(expf(z0 - m) + expf(z1 - m));
    out[g * 2 + 0] = z0 - lse;
    out[g * 2 + 1] = z1 - lse;
}

// ---------------- host launcher ----------------
extern "C" void kernel_launch(void* const* d_in, const int* in_sizes, int n_in,
                              void* d_out, int out_size, void* d_ws, size_t ws_size,
                              hipStream_t stream) {
    const float* x     = (const float*)d_in[0];
    const int*   ei    = (const int*)  d_in[1];   // [2,E]: row0=src, row1=dst
    // d_in[2] = batch (implied by node/64, unused)
    const float* W0    = (const float*)d_in[3];
    const float* b0    = (const float*)d_in[4];
    const float* Ws    = (const float*)d_in[5];
    const float* bs    = (const float*)d_in[6];
    const float* convw = (const float*)d_in[7];
    const float* convb = (const float*)d_in[8];
    const float* lw1   = (const float*)d_in[9];
    const float* lb1   = (const float*)d_in[10];
    const float* lw2   = (const float*)d_in[11];
    const float* lb2   = (const float*)d_in[12];
    const float* lw3   = (const float*)d_in[13];
    const float* lb3   = (const float*)d_in[14];
    float* out = (float*)d_out;

    char* ws = (char*)d_ws;
    size_t off = 0;
    float* dinv = (float*)(ws + off); off += (size_t)N_NODES * 4;              //  0.5 MB
    float* hw   = (float*)(ws + off); off += (size_t)N_NODES * HDIM * 4;       // 64 MB
    float* hbuf = (float*)(ws + off); off += (size_t)N_NODES * HDIM * 4;       // 64 MB
    v2f* gcnWp  = (v2f*)(ws + off);   off += (size_t)4 * HDIM * HDIM * 4;      // 4 packed 128x128
    v2f* fc1p   = (v2f*)(ws + off);   off += (size_t)FC1_K * HDIM * 4;
    v2f* fc2p   = (v2f*)(ws + off);   off += (size_t)HDIM * (HDIM / 2) * 4;
    v2f* wTp    = (v2f*)(ws + off);   off += (size_t)KS_ * HDIM * C1_CH * 4;
    // hw region is dead after the last GCN layer's scatter -> reuse it:
    float* pooled  = hw;                                                       // 31.5 MB
    float* convout = (float*)((char*)pooled  + (size_t)G_GRAPHS * KPOOL * HDIM * 4);
    float* fc1o    = (float*)((char*)convout + (size_t)G_GRAPHS * FC1_K * 4);
    float* fc2o    = (float*)((char*)fc1o    + (size_t)G_GRAPHS * HDIM * 4);

    // 0) pack all weights into pair-row layout (tiny, one-time per call)
    const int gcnWpairs = (HDIM / 2) * HDIM;                                   // 8192 v2f per layer
    pack_b_kernel<<<(gcnWpairs + 255) / 256, 256, 0, stream>>>(W0, gcnWp, HDIM, HDIM);
    for (int i = 0; i < 3; ++i)
        pack_b_kernel<<<(gcnWpairs + 255) / 256, 256, 0, stream>>>(
            Ws + (size_t)i * HDIM * HDIM, gcnWp + (size_t)(i + 1) * gcnWpairs, HDIM, HDIM);
    pack_b_kernel<<<((FC1_K / 2) * HDIM + 255) / 256, 256, 0, stream>>>(lw1, fc1p, FC1_K, HDIM);
    pack_b_kernel<<<((HDIM / 2) * (HDIM / 2) + 255) / 256, 256, 0, stream>>>(lw2, fc2p, HDIM, HDIM / 2);
    convw_pack_kernel<<<((KS_ * HDIM / 2) * C1_CH + 255) / 256, 256, 0, stream>>>(convw, wTp);

    // 1) degrees -> dinv
    deg_init_kernel <<<N_NODES / 256, 256, 0, stream>>>(dinv);
    deg_count_kernel<<<E_EDGES / 256, 256, 0, stream>>>(ei, dinv);
    dinv_kernel     <<<N_NODES / 256, 256, 0, stream>>>(dinv);

    // 2) four GCN layers (ReLU folded into the next GEMM's A-load / sort-pool)
    const int elemBlocks = (N_NODES * HDIM) / 256;                // 65536
    for (int l = 0; l < 4; ++l) {
        const float* b = (l == 0) ? b0 : (bs + (size_t)(l - 1) * HDIM);
        const v2f*  Wp = gcnWp + (size_t)l * gcnWpairs;
        // hw = act(hin) @ W        (65536 tiles / 8 waves per block)
        if (l == 0)
            gemm_wmma_kernel<false><<<8192, 256, 0, stream>>>(x, Wp, nullptr, hw,
                                                              N_NODES, HDIM, HDIM, 0);
        else
            gemm_wmma_kernel<true><<<8192, 256, 0, stream>>>(hbuf, Wp, nullptr, hw,
                                                             N_NODES, HDIM, HDIM, 0);
        // hbuf = hw * dinv^2 + b
        self_bias_kernel<<<elemBlocks, 256, 0, stream>>>(hw, dinv, b, hbuf);
        // hbuf[dst] += hw[src] * dinv[src]*dinv[dst]
        scatter_kernel<<<(E_EDGES * 32) / 256, 256, 0, stream>>>(ei, dinv, hw, hbuf);
    }

    // 3) global sort-pool (ReLU fused; top-30 per graph by last channel)
    sortpool_kernel<<<G_GRAPHS, NPG, 0, stream>>>(hbuf, pooled);

    // 4) conv1d (im2col-free GEMM) + relu, transposed store for reshape order
    conv_gemm_kernel<<<832, 256, 0, stream>>>(pooled, wTp, convb, convout);    // 6656 tiles

    // 5) fc1 (2048x832 @ 832x128, relu) -> fc2 (2048x128 @ 128x64, relu)
    gemm_wmma_kernel<false><<<128, 256, 0, stream>>>(convout, fc1p, lb1, fc1o,
                                                     G_GRAPHS, HDIM, FC1_K, 1);   // 1024 tiles
    gemm_wmma_kernel<false><<<64, 256, 0, stream>>>(fc1o, fc2p, lb2, fc2o,
                                                    G_GRAPHS, HDIM / 2, HDIM, 1); // 512 tiles

    // 6) fc3 + log_softmax
    fc3_lsm_kernel<<<G_GRAPHS / 256, 256, 0, stream>>>(fc2o, lw3, lb3, out);
}